// GCN_GAT_GINE_34780645163298
// MI455X (gfx1250) — compile-verified
//
#include <hip/hip_runtime.h>
#include <stdint.h>

#define NN 20000
#define NE 250000
#define NESL (NE + NN)   // edges incl. self loops
#define NG 200

typedef float v2f __attribute__((ext_vector_type(2)));
typedef float v8f __attribute__((ext_vector_type(8)));

#define GAS __attribute__((address_space(1)))
// explicit global-address-space loads: lower to global_load_* (LOADcnt only,
// no flat/DScnt involvement)
__device__ __forceinline__ v2f  ldg_v2f(const float* p) { return *(const GAS v2f*)p; }
__device__ __forceinline__ float ldg_f (const float* p) { return *(const GAS float*)p; }

// ---------------- order-preserving float<->uint encode for atomic max -------
__device__ __forceinline__ unsigned enc_f(float f) {
  unsigned u = __float_as_uint(f);
  return (u & 0x80000000u) ? ~u : (u | 0x80000000u);
}
__device__ __forceinline__ float dec_f(unsigned u) {
  return (u & 0x80000000u) ? __uint_as_float(u & 0x7fffffffu) : __uint_as_float(~u);
}

__device__ __forceinline__ void edge_sd(const int* __restrict__ ei, int e, int& s, int& d) {
  if (e < NE) { s = ei[e]; d = ei[NE + e]; }
  else        { s = e - NE; d = e - NE; }     // self loop
}

// ---------------- WMMA fp32 GEMM: C[M,N] = A[M,K] @ W[K,N] (+bias)(+relu) ---
// One wave per block. Macro-tile: 80 rows (5 x 16) x 32 cols (2 x 16).
// Requires: M % 80 == 0, N % 32 == 0, K % 4 == 0  (true for every call here).
// Inner loop is branch-free and double-buffered: loads for k+4 are issued
// before the 10 WMMAs for k, so VMEM overlaps the matrix pipe.
#define MT 5
#define NT 2
__global__ __launch_bounds__(32)
void wmma_gemm_kernel(const float* __restrict__ A, const float* __restrict__ W,
                      const float* __restrict__ bias, float* __restrict__ C,
                      int M, int N, int K, int do_relu)
{
  const int lane = threadIdx.x & 31;
  const int half = lane >> 4;       // 0: lanes 0-15 (K=k0,k0+1), 1: lanes 16-31 (K=k0+2,k0+3)
  const int idx  = lane & 15;
  const int col0 = blockIdx.x * (NT * 16);
  const int row0 = blockIdx.y * (MT * 16);

  v8f acc[MT][NT];
#pragma unroll
  for (int i = 0; i < MT; ++i)
#pragma unroll
    for (int j = 0; j < NT; ++j) acc[i][j] = (v8f)0.0f;

  // per-lane streaming pointers
  const float* ap[MT];
#pragma unroll
  for (int mt = 0; mt < MT; ++mt)
    ap[mt] = A + (size_t)(row0 + mt * 16 + idx) * K + 2 * half;
  const float* wp = W + (size_t)(2 * half) * N + (col0 + idx);

  // prologue: fragments for k = 0
  v2f a_cur[MT], b_cur[NT];
#pragma unroll
  for (int mt = 0; mt < MT; ++mt) a_cur[mt] = ldg_v2f(ap[mt]);
#pragma unroll
  for (int nt = 0; nt < NT; ++nt) {
    b_cur[nt].x = ldg_f(wp + nt * 16);
    b_cur[nt].y = ldg_f(wp + nt * 16 + N);
  }

  for (int k0 = 0; k0 < K - 4; k0 += 4) {
    // issue next iteration's loads first
    v2f a_nxt[MT], b_nxt[NT];
#pragma unroll
    for (int mt = 0; mt < MT; ++mt) {
      ap[mt] += 4;
      a_nxt[mt] = ldg_v2f(ap[mt]);
    }
    wp += 4 * (size_t)N;
#pragma unroll
    for (int nt = 0; nt < NT; ++nt) {
      b_nxt[nt].x = ldg_f(wp + nt * 16);
      b_nxt[nt].y = ldg_f(wp + nt * 16 + N);
    }
    // 10 WMMAs on current fragments
#pragma unroll
    for (int nt = 0; nt < NT; ++nt)
#pragma unroll
      for (int mt = 0; mt < MT; ++mt)
        acc[mt][nt] = __builtin_amdgcn_wmma_f32_16x16x4_f32(
            false, a_cur[mt], false, b_cur[nt], (short)0, acc[mt][nt], false, false);
#pragma unroll
    for (int mt = 0; mt < MT; ++mt) a_cur[mt] = a_nxt[mt];
#pragma unroll
    for (int nt = 0; nt < NT; ++nt) b_cur[nt] = b_nxt[nt];
  }
  // epilogue WMMAs for the last k-step
#pragma unroll
  for (int nt = 0; nt < NT; ++nt)
#pragma unroll
    for (int mt = 0; mt < MT; ++mt)
      acc[mt][nt] = __builtin_amdgcn_wmma_f32_16x16x4_f32(
          false, a_cur[mt], false, b_cur[nt], (short)0, acc[mt][nt], false, false);

  // store C (+bias)(+relu): VGPR j of a tile -> row (j + 8*half)
#pragma unroll
  for (int mt = 0; mt < MT; ++mt)
#pragma unroll
    for (int nt = 0; nt < NT; ++nt) {
      const int col = col0 + nt * 16 + idx;
      const float bv = bias ? ldg_f(bias + col) : 0.0f;
#pragma unroll
      for (int j = 0; j < 8; ++j) {
        const int row = row0 + mt * 16 + 8 * half + j;
        float v = acc[mt][nt][j] + bv;
        if (do_relu) v = fmaxf(v, 0.0f);
        C[(size_t)row * N + col] = v;
      }
    }
}

// ---------------- utility kernels ------------------------------------------
__global__ void fill_f32_kernel(float* p, float v, int n) {
  int t = blockIdx.x * blockDim.x + threadIdx.x;
  if (t < n) p[t] = v;
}
__global__ void fill_u32_kernel(unsigned* p, unsigned v, int n) {
  int t = blockIdx.x * blockDim.x + threadIdx.x;
  if (t < n) p[t] = v;
}
__global__ void copy_f32_kernel(const float* __restrict__ s, float* __restrict__ d, int n) {
  int t = blockIdx.x * blockDim.x + threadIdx.x;
  if (t < n) d[t] = s[t];
}

// ---------------- GCN ------------------------------------------------------
__global__ void deg_kernel(const int* __restrict__ ei, float* __restrict__ deg) {
  int e = blockIdx.x * blockDim.x + threadIdx.x;
  if (e < NE) atomicAdd(&deg[ei[NE + e]], 1.0f);
}
__global__ void rsqrt_kernel(float* d, int n) {
  int t = blockIdx.x * blockDim.x + threadIdx.x;
  if (t < n) d[t] = 1.0f / sqrtf(d[t]);   // deg >= 1 (self loop)
}
// out = h * dinv^2 (self loop) + bias
__global__ void gcn_init_kernel(const float* __restrict__ h, const float* __restrict__ dinv,
                                const float* __restrict__ bias, float* __restrict__ out, int F) {
  int t = blockIdx.x * blockDim.x + threadIdx.x;
  if (t >= NN * F) return;
  int i = t / F, c = t % F;
  float di = dinv[i];
  out[t] = h[t] * di * di + bias[c];
}
__global__ void gcn_edge_kernel(const int* __restrict__ ei, const float* __restrict__ h,
                                const float* __restrict__ dinv, float* __restrict__ out, int F) {
  int t = blockIdx.x * blockDim.x + threadIdx.x;
  int ch = F >> 2;
  int e = t / ch;
  if (e >= NE) return;
  int c4 = (t % ch) * 4;
  int s = ei[e], d = ei[NE + e];
  float ns = dinv[s] * dinv[d];
  float4 hv = *(const float4*)(h + (size_t)s * F + c4);
  float* op = out + (size_t)d * F + c4;
  atomicAdd(op + 0, hv.x * ns);
  atomicAdd(op + 1, hv.y * ns);
  atomicAdd(op + 2, hv.z * ns);
  atomicAdd(op + 3, hv.w * ns);
}

// ---------------- BatchNorm(relu(x + bias)) --------------------------------
__global__ void relu_bn_stats_kernel(const float* __restrict__ x, const float* __restrict__ bias,
                                     float* __restrict__ mu, float* __restrict__ rsig, int F) {
  const int c = blockIdx.x;
  const float bv = bias ? bias[c] : 0.0f;
  float s = 0.f, s2 = 0.f;
  for (int r = threadIdx.x; r < NN; r += 256) {
    float v = x[(size_t)r * F + c] + bv;
    v = fmaxf(v, 0.f);
    s += v; s2 += v * v;
  }
  __shared__ float sh[256], sh2[256];
  sh[threadIdx.x] = s; sh2[threadIdx.x] = s2;
  __syncthreads();
  for (int off = 128; off > 0; off >>= 1) {
    if ((int)threadIdx.x < off) {
      sh[threadIdx.x] += sh[threadIdx.x + off];
      sh2[threadIdx.x] += sh2[threadIdx.x + off];
    }
    __syncthreads();
  }
  if (threadIdx.x == 0) {
    float m = sh[0] / (float)NN;
    float var = sh2[0] / (float)NN - m * m;       // biased variance
    mu[c] = m;
    rsig[c] = 1.0f / sqrtf(var + 1e-5f);
  }
}
__global__ void relu_bn_apply_kernel(const float* __restrict__ x, const float* __restrict__ bias,
                                     const float* __restrict__ mu, const float* __restrict__ rsig,
                                     const float* __restrict__ g, const float* __restrict__ b,
                                     float* __restrict__ y, int F) {
  int t = blockIdx.x * blockDim.x + threadIdx.x;
  if (t >= NN * F) return;
  int c = t % F;
  float v = x[t] + (bias ? bias[c] : 0.0f);
  v = fmaxf(v, 0.f);
  y[t] = (v - mu[c]) * rsig[c] * g[c] + b[c];
}

// ---------------- GIN ------------------------------------------------------
__global__ void gin_edge_kernel(const int* __restrict__ ei, const float* __restrict__ x,
                                float* __restrict__ out, int F) {
  int t = blockIdx.x * blockDim.x + threadIdx.x;
  int ch = F >> 2;
  int e = t / ch;
  if (e >= NE) return;
  int c4 = (t % ch) * 4;
  int s = ei[e], d = ei[NE + e];
  float4 hv = *(const float4*)(x + (size_t)s * F + c4);
  float* op = out + (size_t)d * F + c4;
  atomicAdd(op + 0, hv.x);
  atomicAdd(op + 1, hv.y);
  atomicAdd(op + 2, hv.z);
  atomicAdd(op + 3, hv.w);
}

// ---------------- GAT ------------------------------------------------------
__global__ void gat_attn_kernel(const float* __restrict__ h, const float* __restrict__ aws,
                                const float* __restrict__ awd, float* __restrict__ as_o,
                                float* __restrict__ ad_o, int H, int C) {
  int t = blockIdx.x * blockDim.x + threadIdx.x;
  if (t >= NN * H) return;
  int i = t / H, hh = t % H;
  const float* hp = h + (size_t)i * H * C + hh * C;
  const float* ws = aws + hh * C;
  const float* wd = awd + hh * C;
  float ss = 0.f, sd = 0.f;
  for (int c = 0; c < C; ++c) { float v = hp[c]; ss += v * ws[c]; sd += v * wd[c]; }
  as_o[t] = ss; ad_o[t] = sd;
}
__global__ void gat_edge_max_kernel(const int* __restrict__ ei, const float* __restrict__ as,
                                    const float* __restrict__ ad, float* __restrict__ ebuf,
                                    unsigned* __restrict__ menc, int H) {
  int t = blockIdx.x * blockDim.x + threadIdx.x;
  if (t >= NESL * H) return;
  int e = t / H, hh = t % H;
  int s, d; edge_sd(ei, e, s, d);
  float v = as[s * H + hh] + ad[d * H + hh];
  v = (v >= 0.f) ? v : 0.2f * v;                // leaky_relu 0.2
  ebuf[t] = v;
  atomicMax(&menc[d * H + hh], enc_f(v));
}
__global__ void gat_edge_exp_kernel(const int* __restrict__ ei, float* __restrict__ ebuf,
                                    const unsigned* __restrict__ menc, float* __restrict__ ssum,
                                    int H) {
  int t = blockIdx.x * blockDim.x + threadIdx.x;
  if (t >= NESL * H) return;
  int e = t / H, hh = t % H;
  int s, d; edge_sd(ei, e, s, d);
  float ex = expf(ebuf[t] - dec_f(menc[d * H + hh]));
  ebuf[t] = ex;
  atomicAdd(&ssum[d * H + hh], ex);
}
__global__ void gat_edge_agg_kernel(const int* __restrict__ ei, const float* __restrict__ ebuf,
                                    const float* __restrict__ ssum, const float* __restrict__ h,
                                    float* __restrict__ out, int H, int C) {
  int t = blockIdx.x * blockDim.x + threadIdx.x;
  int ch = C >> 2;
  int per = H * ch;
  int e = t / per;
  if (e >= NESL) return;
  int r = t % per;
  int hh = r / ch;
  int c4 = (r % ch) * 4;
  int s, d; edge_sd(ei, e, s, d);
  float alpha = ebuf[e * H + hh] / (ssum[d * H + hh] + 1e-16f);
  float4 hv = *(const float4*)(h + (size_t)s * H * C + hh * C + c4);
  float* op = out + (size_t)d * H * C + hh * C + c4;
  atomicAdd(op + 0, hv.x * alpha);
  atomicAdd(op + 1, hv.y * alpha);
  atomicAdd(op + 2, hv.z * alpha);
  atomicAdd(op + 3, hv.w * alpha);
}

// ---------------- pooling + FC head ----------------------------------------
__global__ void pool_kernel(const float* __restrict__ y, const int* __restrict__ batch,
                            float* __restrict__ cr, int F, int colOff) {
  int t = blockIdx.x * blockDim.x + threadIdx.x;
  if (t >= NN * F) return;
  int i = t / F, c = t % F;
  atomicAdd(&cr[(size_t)batch[i] * 320 + colOff + c], y[t]);
}
__global__ void fc_kernel(const float* __restrict__ A, const float* __restrict__ W,
                          const float* __restrict__ bias, float* __restrict__ C,
                          int M, int N, int K, int do_relu) {
  int t = blockIdx.x * blockDim.x + threadIdx.x;
  if (t >= M * N) return;
  int m = t / N, n = t % N;
  float s = bias[n];
  for (int k = 0; k < K; ++k) s += A[m * K + k] * W[k * N + n];
  if (do_relu) s = fmaxf(s, 0.f);
  C[t] = s;
}

// ---------------------------------------------------------------------------
extern "C" void kernel_launch(void* const* d_in, const int* in_sizes, int n_in,
                              void* d_out, int out_size, void* d_ws, size_t ws_size,
                              hipStream_t stream) {
  (void)in_sizes; (void)n_in; (void)out_size; (void)ws_size;
  auto P = [&](int i) { return (const float*)d_in[i]; };

  const float* x     = P(0);
  const int*   ei    = (const int*)d_in[1];
  const int*   batch = (const int*)d_in[2];
  // params (setup_inputs dict insertion order)
  const float *conv1_w = P(3),  *conv1_b = P(4),  *bn01_g = P(5),  *bn01_b = P(6);
  const float *conv2_w = P(7),  *conv2_b = P(8),  *bn02_g = P(9),  *bn02_b = P(10);
  const float *conv3_w = P(11), *conv3_b = P(12), *bn03_g = P(13), *bn03_b = P(14);
  const float *gat1_w = P(15), *gat1_as = P(16), *gat1_ad = P(17), *gat1_b = P(18);
  const float *bn11_g = P(19), *bn11_b = P(20);
  const float *gat2_w = P(21), *gat2_as = P(22), *gat2_ad = P(23), *gat2_b = P(24);
  const float *bn12_g = P(25), *bn12_b = P(26);
  const float *gat3_w = P(27), *gat3_as = P(28), *gat3_ad = P(29), *gat3_b = P(30);
  const float *bn13_g = P(31), *bn13_b = P(32);
  const float *gin1_w1 = P(33), *gin1_b1 = P(34), *gin1_w2 = P(35), *gin1_b2 = P(36);
  const float *bn21_g = P(37), *bn21_b = P(38);
  const float *gin2_w1 = P(39), *gin2_b1 = P(40), *gin2_w2 = P(41), *gin2_b2 = P(42);
  const float *bn22_g = P(43), *bn22_b = P(44);
  const float *gin3_w1 = P(45), *gin3_b1 = P(46), *gin3_w2 = P(47), *gin3_b2 = P(48);
  const float *bn23_g = P(49), *bn23_b = P(50);
  const float *fc1_w = P(51), *fc1_b = P(52);
  const float *fc2_w = P(53), *fc2_b = P(54);
  const float *fc3_w = P(55), *fc3_b = P(56);

  // ------ workspace arena (~175 MB) ------
  char* w = (char*)d_ws;
  const size_t BIG = (size_t)NN * 1024 * 4;            // 81,920,000 B
  float*    big0 = (float*)w;   w += BIG;
  float*    big1 = (float*)w;   w += BIG;
  float*    ebuf = (float*)w;   w += (size_t)NESL * 8 * 4;
  float*    as_b = (float*)w;   w += (size_t)NN * 8 * 4;
  float*    ad_b = (float*)w;   w += (size_t)NN * 8 * 4;
  unsigned* menc = (unsigned*)w; w += (size_t)NN * 8 * 4;
  float*    ssum = (float*)w;   w += (size_t)NN * 8 * 4;
  float*    dinv = (float*)w;   w += (size_t)NN * 4;
  float*    mu   = (float*)w;   w += 1024 * 4;
  float*    rsig = (float*)w;   w += 1024 * 4;
  float*    cr   = (float*)w;   w += (size_t)NG * 320 * 4;
  float*    fc1o = (float*)w;   w += (size_t)NG * 128 * 4;
  float*    fc2o = (float*)w;   w += (size_t)NG * 64 * 4;

  auto gemm = [&](const float* A, const float* Wm, const float* bias, float* C,
                  int M, int N, int K, int relu) {
    dim3 g((unsigned)(N / 32), (unsigned)(M / 80));   // exact: M%80==0, N%32==0
    wmma_gemm_kernel<<<g, dim3(32), 0, stream>>>(A, Wm, bias, C, M, N, K, relu);
  };
  auto fill = [&](float* p, float v, int n) {
    fill_f32_kernel<<<(n + 255) / 256, 256, 0, stream>>>(p, v, n);
  };
  auto fillu = [&](unsigned* p, unsigned v, int n) {
    fill_u32_kernel<<<(n + 255) / 256, 256, 0, stream>>>(p, v, n);
  };
  auto copyf = [&](const float* s, float* d, int n) {
    copy_f32_kernel<<<(n + 255) / 256, 256, 0, stream>>>(s, d, n);
  };
  auto relu_bn = [&](float* xb, const float* bias, const float* g, const float* b, int F) {
    relu_bn_stats_kernel<<<F, 256, 0, stream>>>(xb, bias, mu, rsig, F);
    int n = NN * F;
    relu_bn_apply_kernel<<<(n + 255) / 256, 256, 0, stream>>>(xb, bias, mu, rsig, g, b, xb, F);
  };

  fill(cr, 0.f, NG * 320);

  // ================= GCN tower =================
  fill(dinv, 1.0f, NN);                                    // self-loop degree
  deg_kernel<<<(NE + 255) / 256, 256, 0, stream>>>(ei, dinv);
  rsqrt_kernel<<<(NN + 255) / 256, 256, 0, stream>>>(dinv, NN);

  auto gcn_layer = [&](const float* hin, const float* Wm, const float* bias, int Fin, int Fout,
                       const float* g, const float* b) {
    gemm(hin, Wm, nullptr, big0, NN, Fout, Fin, 0);
    int n = NN * Fout;
    gcn_init_kernel<<<(n + 255) / 256, 256, 0, stream>>>(big0, dinv, bias, big1, Fout);
    int ne = NE * (Fout / 4);
    gcn_edge_kernel<<<(ne + 255) / 256, 256, 0, stream>>>(ei, big0, dinv, big1, Fout);
    relu_bn(big1, nullptr, g, b, Fout);
  };
  gcn_layer(x,    conv1_w, conv1_b, 64,  128, bn01_g, bn01_b);
  gcn_layer(big1, conv2_w, conv2_b, 128, 64,  bn02_g, bn02_b);
  gcn_layer(big1, conv3_w, conv3_b, 64,  32,  bn03_g, bn03_b);
  pool_kernel<<<(NN * 32 + 255) / 256, 256, 0, stream>>>(big1, batch, cr, 32, 0);

  // ================= GIN tower =================
  auto gin_layer = [&](const float* zin, int Fin, int Fh, int Fo,
                       const float* w1, const float* b1, const float* w2, const float* b2,
                       const float* g, const float* bb,
                       float* agg, float* tmp, float* outb) {
    copyf(zin, agg, NN * Fin);                              // h = x + sum(x[src])
    int ne = NE * (Fin / 4);
    gin_edge_kernel<<<(ne + 255) / 256, 256, 0, stream>>>(ei, zin, agg, Fin);
    gemm(agg, w1, b1, tmp, NN, Fh, Fin, 1);                 // relu(h@w1+b1)
    gemm(tmp, w2, b2, outb, NN, Fo, Fh, 0);                 // @w2+b2
    relu_bn(outb, nullptr, g, bb, Fo);
  };
  gin_layer(x,    64,  128, 128, gin1_w1, gin1_b1, gin1_w2, gin1_b2, bn21_g, bn21_b,
            big0, big1, big0);                              // z1 -> big0
  gin_layer(big0, 128, 64,  64,  gin2_w1, gin2_b1, gin2_w2, gin2_b2, bn22_g, bn22_b,
            big1, big0, big1);                              // z2 -> big1
  gin_layer(big1, 64,  32,  32,  gin3_w1, gin3_b1, gin3_w2, gin3_b2, bn23_g, bn23_b,
            big0, big1, big0);                              // z3 -> big0
  pool_kernel<<<(NN * 32 + 255) / 256, 256, 0, stream>>>(big0, batch, cr, 32, 288);

  // ================= GAT tower =================
  auto gat_layer = [&](const float* h, const float* aws, const float* awd, float* outb, int C) {
    const int H = 8;
    int nAttn = NN * H;
    gat_attn_kernel<<<(nAttn + 255) / 256, 256, 0, stream>>>(h, aws, awd, as_b, ad_b, H, C);
    fillu(menc, 0u, NN * H);
    fill(ssum, 0.f, NN * H);
    fill(outb, 0.f, NN * H * C);
    int nEH = NESL * H;
    gat_edge_max_kernel<<<(nEH + 255) / 256, 256, 0, stream>>>(ei, as_b, ad_b, ebuf, menc, H);
    gat_edge_exp_kernel<<<(nEH + 255) / 256, 256, 0, stream>>>(ei, ebuf, menc, ssum, H);
    int nAgg = NESL * H * (C / 4);
    gat_edge_agg_kernel<<<(nAgg + 255) / 256, 256, 0, stream>>>(ei, ebuf, ssum, h, outb, H, C);
  };

  // layer 1: H=8, C=128 (h1=big0, out=big1)
  gemm(x, gat1_w, nullptr, big0, NN, 1024, 64, 0);
  gat_layer(big0, gat1_as, gat1_ad, big1, 128);
  relu_bn(big1, gat1_b, bn11_g, bn11_b, 1024);              // y1 = big1

  // layer 2: H=8, C=64 (h2=big0[:512], out=big0[512:])
  gemm(big1, gat2_w, nullptr, big0, NN, 512, 1024, 0);
  float* out2 = big0 + (size_t)NN * 512;
  gat_layer(big0, gat2_as, gat2_ad, out2, 64);
  relu_bn(out2, gat2_b, bn12_g, bn12_b, 512);               // y2 = out2

  // layer 3: H=8, C=32 (h3=big1[:256], out=big1[256:])
  gemm(out2, gat3_w, nullptr, big1, NN, 256, 512, 0);
  float* out3 = big1 + (size_t)NN * 256;
  gat_layer(big1, gat3_as, gat3_ad, out3, 32);
  relu_bn(out3, gat3_b, bn13_g, bn13_b, 256);
  pool_kernel<<<(NN * 256 + 255) / 256, 256, 0, stream>>>(out3, batch, cr, 256, 32);

  // ================= FC head =================
  fc_kernel<<<(NG * 128 + 255) / 256, 256, 0, stream>>>(cr,   fc1_w, fc1_b, fc1o, NG, 128, 320, 1);
  fc_kernel<<<(NG * 64 + 255) / 256, 256, 0, stream>>>(fc1o, fc2_w, fc2_b, fc2o, NG, 64, 128, 1);
  fc_kernel<<<(NG * 2 + 255) / 256, 256, 0, stream>>>(fc2o, fc3_w, fc3_b, (float*)d_out, NG, 2, 64, 0);
}